// GCNLSTMCell_31920196943931
// MI455X (gfx1250) — compile-verified
//
#include <hip/hip_runtime.h>
#include <math.h>

// Problem constants (match reference setup_inputs)
constexpr int Bb = 8;       // batch (== LSTM time steps == H)
constexpr int Nn = 10000;   // nodes
constexpr int Hh = 8;       // hidden
constexpr int Ee = 160000;  // edges
constexpr int Mrows = Bb * Nn;      // 80000 rows of x flattened
constexpr int NS = Nn * 64;         // 640000
constexpr int NTILES = Nn / 16;     // 625

typedef __attribute__((ext_vector_type(2))) float v2f;
typedef __attribute__((ext_vector_type(8))) float v8f;
typedef __attribute__((ext_vector_type(4))) unsigned int v4u;
typedef __attribute__((ext_vector_type(8))) int v8i;
typedef __attribute__((ext_vector_type(4))) int v4i;

// D(16x16,f32) = A(16x4,f32) * B(4x16,f32) + C   -- wave32
__device__ __forceinline__ v8f wmma4(v2f a, v2f b, v8f c) {
  return __builtin_amdgcn_wmma_f32_16x16x4_f32(
      /*neg_a=*/false, a, /*neg_b=*/false, b,
      /*c_mod=*/(short)0, c, /*reuse_a=*/false, /*reuse_b=*/false);
}

__device__ __forceinline__ float sigf(float x) {
  return 1.0f / (1.0f + __expf(-x));
}

// low 32 bits of a generic LDS address == LDS byte offset (ISA aperture rule)
__device__ __forceinline__ unsigned lds_off(const void* p) {
  return (unsigned)(unsigned long long)p;
}

// ---------------- degree / normalization ----------------
__global__ void init_deg(float* deg) {
  int i = blockIdx.x * blockDim.x + threadIdx.x;
  if (i < Nn) deg[i] = 1.0f;
}

__global__ void accum_deg(const int* __restrict__ dst,
                          const float* __restrict__ w, float* deg) {
  int e = blockIdx.x * blockDim.x + threadIdx.x;
  if (e < Ee) atomicAdd(&deg[dst[e]], w[e]);
}

__global__ void finalize_dis(float* deg) {  // in-place -> dis
  int i = blockIdx.x * blockDim.x + threadIdx.x;
  if (i < Nn) deg[i] = rsqrtf(deg[i]);
}

// ---------------- GCN GEMM: xw = x * W^T ; y = dis^2*xw + b ----------------
// grid 625 x 256 (8 waves); wave = 16 rows x 64 cols, K=64.
// W staged by the Tensor Data Mover (row-padded 64->68 by TDM pad logic);
// per-wave A tiles staged by async global->LDS B128 copies.
__global__ void gcn_gemm(const float* __restrict__ x,
                         const float* __restrict__ Wg,
                         const float* __restrict__ bg,
                         const float* __restrict__ dis,
                         float* __restrict__ xw, float* __restrict__ y) {
  // stride-68 rows: 272B (16B aligned) and bank-conflict-free
  __shared__ __attribute__((aligned(16))) float smem[64 * 68 + 8 * 16 * 68];
  float* Wl = smem;  // Wl[o*68 + k]

  const int tid = threadIdx.x;
  const int wid = tid >> 5;
  const int lane = tid & 31;
  const int half = lane >> 4;
  const int lm = lane & 15;
  const int rowbase = blockIdx.x * 128 + wid * 16;
  float* xsw = smem + 64 * 68 + wid * (16 * 68);  // this wave's A tile

  // --- TDM: DMA the 64x64 f32 W tile into LDS, pad 4 DWORDs per 64 DWORDs ---
  if (wid == 0) {
    const unsigned lds0 = lds_off(&smem[0]);
    const unsigned long long ga = (unsigned long long)Wg;
    v4u g0;
    g0[0] = 1u;                                   // count=1, user descriptor
    g0[1] = lds0;                                 // lds_addr
    g0[2] = (unsigned)ga;                         // global_addr[31:0]
    g0[3] = (unsigned)(ga >> 32) | (2u << 30);    // global_addr[56:32] | type=2
    v8i g1;
    g1[0] = (int)((2u << 16)      // data_size = 4B
                  | (1u << 20)    // pad_enable
                  | (5u << 22)    // pad_interval: every 64 DWORDs
                  | (3u << 25));  // pad_amount: 4 DWORDs
    g1[1] = (int)(64u << 16);     // tensor_dim0 = 64
    g1[2] = (int)(64u << 16);     // tensor_dim1 = 64
    g1[3] = (int)(64u << 16);     // tile_dim0 = 64
    g1[4] = 64;                   // tile_dim1 = 64
    g1[5] = 64;                   // tensor_dim0_stride = 64
    g1[6] = 0;
    g1[7] = 0;
    v4i z4 = {0, 0, 0, 0};
#if __clang_major__ >= 23
    v8i z8 = {0, 0, 0, 0, 0, 0, 0, 0};
    __builtin_amdgcn_tensor_load_to_lds(g0, g1, z4, z4, z8, 0);
#else
    __builtin_amdgcn_tensor_load_to_lds(g0, g1, z4, z4, 0);
#endif
    __builtin_amdgcn_s_wait_tensorcnt(0);
  }

  // --- async-copy this wave's 16x64 A tile into LDS (stride-68 rows) ---
  {
    const unsigned ldsx = lds_off(xsw);
    for (int i = lane; i < 256; i += 32) {  // 256 x 16B chunks
      const int r = i >> 4, cv = i & 15;
      const unsigned ldst = ldsx + (unsigned)(r * 272 + cv * 16);
      const unsigned goff = (unsigned)(rowbase * 256 + i * 16);
      asm volatile("global_load_async_to_lds_b128 %0, %1, %2"
                   :: "v"(ldst), "v"(goff), "s"(x)
                   : "memory");
    }
    asm volatile("s_wait_asynccnt 0" ::: "memory");
  }
  __syncthreads();

  v8f acc[4] = {};
  for (int k0 = 0; k0 < 64; k0 += 4) {
    const int kb = k0 + half * 2;
    v2f a;
    a.x = xsw[lm * 68 + kb];
    a.y = xsw[lm * 68 + kb + 1];
#pragma unroll
    for (int ot = 0; ot < 4; ++ot) {
      const int o = ot * 16 + lm;
      v2f b;
      b.x = Wl[o * 68 + kb];
      b.y = Wl[o * 68 + kb + 1];
      acc[ot] = wmma4(a, b, acc[ot]);
    }
  }

  // epilogue: 16-row tiles never cross a batch boundary (10000 % 16 == 0)
  const int nbase = rowbase % Nn;
  float bgv[4];
#pragma unroll
  for (int ot = 0; ot < 4; ++ot) bgv[ot] = bg[ot * 16 + lm];
#pragma unroll
  for (int vg = 0; vg < 8; ++vg) {
    const int rloc = vg + 8 * half;
    const int r = rowbase + rloc;
    const float di = dis[nbase + rloc];
    const float d2 = di * di;
#pragma unroll
    for (int ot = 0; ot < 4; ++ot) {
      const int o = ot * 16 + lm;
      const float val = acc[ot][vg];
      xw[r * 64 + o] = val;
      y[r * 64 + o] = d2 * val + bgv[ot];
    }
  }
}

// ---------------- edge scatter: y[:,dst,:] += norm * xw[:,src,:] ----------
__global__ void edge_scatter(const int* __restrict__ src,
                             const int* __restrict__ dst,
                             const float* __restrict__ w,
                             const float* __restrict__ dis,
                             const float* __restrict__ xw,
                             float* __restrict__ y) {
  const int wg = (blockIdx.x * blockDim.x + threadIdx.x) >> 5;
  const int lane = threadIdx.x & 31;
  if (wg >= Ee) return;
  const int s = src[wg];
  const int d = dst[wg];
  const float norm = dis[s] * dis[d] * w[wg];
  const float* xr = xw + s * 64;
  float* yr = y + d * 64;
#pragma unroll
  for (int i = 0; i < 16; ++i) {
    const int idx = i * 32 + lane;
    const int b = idx >> 6;
    const int cofs = (idx & 63) + b * NS;
    atomicAdd(&yr[cofs], norm * xr[cofs]);
  }
}

// ---------------- LSTM ----------------
__global__ void zero_hc(float* h, float* c) {
  int i = blockIdx.x * blockDim.x + threadIdx.x;
  if (i < Nn * Hh) { h[i] = 0.0f; c[i] = 0.0f; }
}

// one wave = 16 nodes x 32 gate cols
__global__ void lstm_step(const float* __restrict__ y,
                          const float* __restrict__ Wih,
                          const float* __restrict__ Whh,
                          const float* __restrict__ bih,
                          const float* __restrict__ bhh,
                          float* __restrict__ h, float* __restrict__ c,
                          float* __restrict__ v, int t) {
  __shared__ float gbuf[8][16][33];
  const int tid = threadIdx.x;
  const int wid = tid >> 5;
  const int lane = tid & 31;
  const int half = lane >> 4;
  const int lm = lane & 15;

  int tile = blockIdx.x * 8 + wid;
  const bool valid = tile < NTILES;
  if (!valid) tile = NTILES - 1;  // compute garbage, never write
  const int n0 = tile * 16;
  const int n = n0 + lm;

  const float* yt = y + t * NS;  // x2[t][n][s] = yt[s*N + n]

  v8f acc[2] = {};
  for (int k0 = 0; k0 < 64; k0 += 4) {
    const int kb = k0 + half * 2;
    v2f a;
    a.x = yt[kb * Nn + n];
    a.y = yt[(kb + 1) * Nn + n];
#pragma unroll
    for (int ot = 0; ot < 2; ++ot) {
      const int o = ot * 16 + lm;
      v2f b;
      b.x = Wih[o * 64 + kb];
      b.y = Wih[o * 64 + kb + 1];
      acc[ot] = wmma4(a, b, acc[ot]);
    }
  }
  // recurrent: K = 8
#pragma unroll
  for (int k0 = 0; k0 < 8; k0 += 4) {
    const int kb = k0 + half * 2;
    v2f a;
    a.x = h[n * 8 + kb];
    a.y = h[n * 8 + kb + 1];
#pragma unroll
    for (int ot = 0; ot < 2; ++ot) {
      const int o = ot * 16 + lm;
      v2f b;
      b.x = Whh[o * 8 + kb];
      b.y = Whh[o * 8 + kb + 1];
      acc[ot] = wmma4(a, b, acc[ot]);
    }
  }

  // re-layout gates through LDS (add bias here)
#pragma unroll
  for (int ot = 0; ot < 2; ++ot) {
    const int o = ot * 16 + lm;
    const float bs = bih[o] + bhh[o];
#pragma unroll
    for (int vg = 0; vg < 8; ++vg)
      gbuf[wid][vg + 8 * half][o] = acc[ot][vg] + bs;
  }
  __syncthreads();

  // elementwise LSTM update: 128 (node,hidden) items per wave, 4 per lane
#pragma unroll
  for (int i = 0; i < 4; ++i) {
    const int item = lane + 32 * i;
    const int rr = item >> 3;
    const int j = item & 7;
    const float gi = gbuf[wid][rr][j];
    const float gf = gbuf[wid][rr][8 + j];
    const float gg = gbuf[wid][rr][16 + j];
    const float go = gbuf[wid][rr][24 + j];
    const int n2 = n0 + rr;
    const float cold = c[n2 * 8 + j];
    const float cn = sigf(gf) * cold + sigf(gi) * tanhf(gg);
    const float hn = sigf(go) * tanhf(cn);
    if (valid) {
      c[n2 * 8 + j] = cn;
      h[n2 * 8 + j] = hn;
      if (j == 7) v[t * Nn + n2] = hn;
    }
  }
}

// ---------------- FC head + broadcast output ----------------
__global__ void fc_out(const float* __restrict__ v,
                       const float* __restrict__ Wfc,
                       const float* __restrict__ bfc,
                       float* __restrict__ out) {
  const int tid = threadIdx.x;
  const int wid = tid >> 5;
  const int lane = tid & 31;
  const int tile = blockIdx.x * 8 + wid;
  if (tile >= NTILES) return;  // whole-wave exit; EXEC stays all-1 for wmma
  const int half = lane >> 4;
  const int lm = lane & 15;
  const int n0 = tile * 16;
  const int n = n0 + lm;

  v8f acc[4] = {};
#pragma unroll
  for (int k0 = 0; k0 < 8; k0 += 4) {
    const int kb = k0 + half * 2;
    v2f a;
    a.x = v[kb * Nn + n];
    a.y = v[(kb + 1) * Nn + n];
#pragma unroll
    for (int ot = 0; ot < 4; ++ot) {
      const int o = ot * 16 + lm;
      v2f b;
      b.x = Wfc[o * 8 + kb];
      b.y = Wfc[o * 8 + kb + 1];
      acc[ot] = wmma4(a, b, acc[ot]);
    }
  }

#pragma unroll
  for (int ot = 0; ot < 4; ++ot) {
    const int o = ot * 16 + lm;
    const float bv = bfc[o];
#pragma unroll
    for (int vg = 0; vg < 8; ++vg) {
      const int r = n0 + vg + 8 * half;
      float val = acc[ot][vg] + bv;
      val = fmaxf(val, 0.0f);
#pragma unroll
      for (int b = 0; b < Bb; ++b) out[b * NS + r * 64 + o] = val;
    }
  }
}

// ---------------- launch ----------------
extern "C" void kernel_launch(void* const* d_in, const int* in_sizes, int n_in,
                              void* d_out, int out_size, void* d_ws,
                              size_t ws_size, hipStream_t stream) {
  (void)in_sizes; (void)n_in; (void)out_size; (void)ws_size;
  const float* x   = (const float*)d_in[0];
  const int*   ei  = (const int*)d_in[1];     // (2, E): src = ei, dst = ei+E
  const float* ew  = (const float*)d_in[2];
  const float* Wg  = (const float*)d_in[3];
  const float* bg  = (const float*)d_in[4];
  const float* Wih = (const float*)d_in[5];
  const float* Whh = (const float*)d_in[6];
  const float* bih = (const float*)d_in[7];
  const float* bhh = (const float*)d_in[8];
  const float* Wfc = (const float*)d_in[9];
  const float* bfc = (const float*)d_in[10];
  float* out = (float*)d_out;

  // workspace layout (floats): ~42 MB total
  float* ws  = (float*)d_ws;
  float* dis = ws;                    // 10000 (deg -> dis in place)
  float* xw  = ws + 10240;            // 5,120,000
  float* y   = xw + (size_t)Bb * NS;  // 5,120,000
  float* h   = y + (size_t)Bb * NS;   // 80,000
  float* c   = h + Nn * Hh;           // 80,000
  float* v   = c + Nn * Hh;           // 80,000  (v[t*N + n])

  init_deg<<<(Nn + 255) / 256, 256, 0, stream>>>(dis);
  accum_deg<<<(Ee + 255) / 256, 256, 0, stream>>>(ei + Ee, ew, dis);
  finalize_dis<<<(Nn + 255) / 256, 256, 0, stream>>>(dis);

  gcn_gemm<<<Mrows / 128, 256, 0, stream>>>(x, Wg, bg, dis, xw, y);
  edge_scatter<<<(Ee * 32) / 256, 256, 0, stream>>>(ei, ei + Ee, ew, dis, xw, y);

  zero_hc<<<(Nn * Hh + 255) / 256, 256, 0, stream>>>(h, c);
  for (int t = 0; t < Bb; ++t)
    lstm_step<<<(NTILES + 7) / 8, 256, 0, stream>>>(y, Wih, Whh, bih, bhh, h, c, v, t);

  fc_out<<<(NTILES + 7) / 8, 256, 0, stream>>>(v, Wfc, bfc, out);
}